// SimpleGCN_36618891166257
// MI455X (gfx1250) — compile-verified
//
#include <hip/hip_runtime.h>

typedef __attribute__((ext_vector_type(2))) float v2f;
typedef __attribute__((ext_vector_type(8))) float v8f;

#define GCN_IN_DIM  128
#define GCN_HID_DIM 128
#define GCN_OUT_DIM 64

// ---------------------------------------------------------------------------
// Degree / normalization kernels
// ---------------------------------------------------------------------------
__global__ void deg_init_kernel(float* __restrict__ dinv, int n) {
  int i = blockIdx.x * blockDim.x + threadIdx.x;
  if (i < n) dinv[i] = 1.0f;  // self-loop contributes 1 to every node's degree
}

__global__ void deg_count_kernel(const int* __restrict__ dst,
                                 float* __restrict__ dinv, int e) {
  int i = blockIdx.x * blockDim.x + threadIdx.x;
  if (i < e) atomicAdd(&dinv[dst[i]], 1.0f);
}

__global__ void deg_rsqrt_kernel(float* __restrict__ dinv, int n) {
  int i = blockIdx.x * blockDim.x + threadIdx.x;
  if (i < n) dinv[i] = rsqrtf(dinv[i]);  // deg >= 1 always (self loops)
}

// ---------------------------------------------------------------------------
// Dense GEMM  T[M,N] = X[M,K] @ W[K,N]  via V_WMMA_F32_16X16X4_F32 (wave32).
// Each wave computes a 16(M) x 64(N) slab: 4 accumulators of v8f.
// Requires M%16==0, K%4==0, N%64==0 (true here: M=50000, K=128, N=128/64).
// A-frag (16x4 f32): lanes 0-15 rows, lane>=16 holds K+2 pair.
// B-frag (4x16 f32): mirrored striping across lanes within VGPR.
// C/D (16x16 f32): VGPR r -> row m0+r (lanes 0-15) / m0+8+r (lanes 16-31).
// ---------------------------------------------------------------------------
__launch_bounds__(128)
__global__ void gemm_wmma_f32_kernel(const float* __restrict__ X,
                                     const float* __restrict__ W,
                                     float* __restrict__ T,
                                     int M, int K, int N) {
  const int lane   = threadIdx.x & 31;
  const int wave   = threadIdx.x >> 5;
  const int mtiles = M >> 4;
  const int ntiles = N >> 6;
  const int tile   = blockIdx.x * 4 + wave;
  if (tile >= mtiles * ntiles) return;  // wave-uniform exit: EXEC stays all-1s

  const int m0    = (tile % mtiles) << 4;
  const int n0    = (tile / mtiles) << 6;
  const int row   = m0 + (lane & 15);
  const int khalf = (lane >> 4) << 1;  // 0 for lanes 0-15, 2 for lanes 16-31
  const int colb  = lane & 15;

  v8f acc0 = {}, acc1 = {}, acc2 = {}, acc3 = {};
  const float* __restrict__ xrow = X + (long)row * K + khalf;

  for (int k0 = 0; k0 < K; k0 += 4) {
    v2f a;
    a.x = xrow[k0];
    a.y = xrow[k0 + 1];
    const float* __restrict__ bp = W + (long)(k0 + khalf) * N + n0 + colb;
    v2f b0; b0.x = bp[0];  b0.y = bp[N];
    v2f b1; b1.x = bp[16]; b1.y = bp[N + 16];
    v2f b2; b2.x = bp[32]; b2.y = bp[N + 32];
    v2f b3; b3.x = bp[48]; b3.y = bp[N + 48];
    acc0 = __builtin_amdgcn_wmma_f32_16x16x4_f32(false, a, false, b0, (short)0, acc0, false, false);
    acc1 = __builtin_amdgcn_wmma_f32_16x16x4_f32(false, a, false, b1, (short)0, acc1, false, false);
    acc2 = __builtin_amdgcn_wmma_f32_16x16x4_f32(false, a, false, b2, (short)0, acc2, false, false);
    acc3 = __builtin_amdgcn_wmma_f32_16x16x4_f32(false, a, false, b3, (short)0, acc3, false, false);
  }

  const int r0 = m0 + ((lane >> 4) << 3);
  float* __restrict__ out = T + (long)r0 * N + n0 + colb;
#pragma unroll
  for (int r = 0; r < 8; ++r) {
    out[(long)r * N +  0] = acc0[r];
    out[(long)r * N + 16] = acc1[r];
    out[(long)r * N + 32] = acc2[r];
    out[(long)r * N + 48] = acc3[r];
  }
}

// ---------------------------------------------------------------------------
// agg[v,:] = T[v,:] * dinv[v]^2  (self-loop term; also zero-initializes agg)
// ---------------------------------------------------------------------------
__global__ void self_init_kernel(const float* __restrict__ T,
                                 const float* __restrict__ dinv,
                                 float* __restrict__ agg, int n, int D) {
  int i = blockIdx.x * blockDim.x + threadIdx.x;
  if (i < n * D) {
    int v = i / D;
    float di = dinv[v];
    agg[i] = T[i] * (di * di);
  }
}

// ---------------------------------------------------------------------------
// For each edge (s -> d): agg[d,:] += T[s,:] * dinv[s]*dinv[d]
// D/4 threads per edge, float4 gather, f32 global atomics (no return).
// ---------------------------------------------------------------------------
template <int D>
__global__ void edge_scatter_kernel(const int* __restrict__ src,
                                    const int* __restrict__ dst,
                                    const float* __restrict__ dinv,
                                    const float* __restrict__ T,
                                    float* __restrict__ agg, int E) {
  constexpr int TPE = D / 4;
  int gid = blockIdx.x * blockDim.x + threadIdx.x;
  int e = gid / TPE;
  if (e >= E) return;
  int c = (gid % TPE) * 4;
  int s = src[e];
  int d = dst[e];
  float w = dinv[s] * dinv[d];
  const float4 tv = *(const float4*)(T + (long)s * D + c);
  float* __restrict__ ap = agg + (long)d * D + c;
  atomicAdd(ap + 0, tv.x * w);
  atomicAdd(ap + 1, tv.y * w);
  atomicAdd(ap + 2, tv.z * w);
  atomicAdd(ap + 3, tv.w * w);
}

// ---------------------------------------------------------------------------
// h[i] = (relu?) (h[i] + b[i % D])    (in place)
// ---------------------------------------------------------------------------
template <bool RELU>
__global__ void bias_act_kernel(float* __restrict__ h,
                                const float* __restrict__ b, int n, int D) {
  int i = blockIdx.x * blockDim.x + threadIdx.x;
  if (i < n * D) {
    float v = h[i] + b[i % D];
    h[i] = RELU ? fmaxf(v, 0.0f) : v;
  }
}

// ---------------------------------------------------------------------------
extern "C" void kernel_launch(void* const* d_in, const int* in_sizes, int n_in,
                              void* d_out, int out_size, void* d_ws, size_t ws_size,
                              hipStream_t stream) {
  (void)n_in; (void)out_size; (void)ws_size;

  const float* x  = (const float*)d_in[0];
  const int*   ei = (const int*)d_in[1];
  const float* W1 = (const float*)d_in[2];
  const float* b1 = (const float*)d_in[3];
  const float* W2 = (const float*)d_in[4];
  const float* b2 = (const float*)d_in[5];

  const int N = in_sizes[0] / GCN_IN_DIM;  // 50000
  const int E = in_sizes[1] / 2;           // 800000
  const int* src = ei;       // edge_index[0]
  const int* dst = ei + E;   // edge_index[1]

  float* out = (float*)d_out;
  float* h2  = out;                          // [N, OUT_DIM]
  float* h1  = out + (long)N * GCN_OUT_DIM;  // [N, HID_DIM]

  char*  ws   = (char*)d_ws;
  float* dinv = (float*)ws;
  float* T    = (float*)(ws + (((long)N * sizeof(float) + 255) / 256) * 256);

  const int TB = 256;

  // --- symmetric normalization: dinv = rsqrt(1 + in-degree) ---
  deg_init_kernel<<<(N + TB - 1) / TB, TB, 0, stream>>>(dinv, N);
  deg_count_kernel<<<(E + TB - 1) / TB, TB, 0, stream>>>(dst, dinv, E);
  deg_rsqrt_kernel<<<(N + TB - 1) / TB, TB, 0, stream>>>(dinv, N);

  // --- layer 1: h1 = relu(norm-agg(x @ W1) + b1) ---
  {
    const int tiles = (N / 16) * (GCN_HID_DIM / 64);
    gemm_wmma_f32_kernel<<<(tiles + 3) / 4, 128, 0, stream>>>(
        x, W1, T, N, GCN_IN_DIM, GCN_HID_DIM);
    const int tot = N * GCN_HID_DIM;
    self_init_kernel<<<(tot + TB - 1) / TB, TB, 0, stream>>>(T, dinv, h1, N, GCN_HID_DIM);
    const long thr = (long)E * (GCN_HID_DIM / 4);
    edge_scatter_kernel<GCN_HID_DIM><<<(int)((thr + TB - 1) / TB), TB, 0, stream>>>(
        src, dst, dinv, T, h1, E);
    bias_act_kernel<true><<<(tot + TB - 1) / TB, TB, 0, stream>>>(h1, b1, N, GCN_HID_DIM);
  }

  // --- layer 2: h2 = norm-agg(h1 @ W2) + b2 ---
  {
    const int tiles = (N / 16) * (GCN_OUT_DIM / 64);
    gemm_wmma_f32_kernel<<<(tiles + 3) / 4, 128, 0, stream>>>(
        h1, W2, T, N, GCN_HID_DIM, GCN_OUT_DIM);
    const int tot = N * GCN_OUT_DIM;
    self_init_kernel<<<(tot + TB - 1) / TB, TB, 0, stream>>>(T, dinv, h2, N, GCN_OUT_DIM);
    const long thr = (long)E * (GCN_OUT_DIM / 4);
    edge_scatter_kernel<GCN_OUT_DIM><<<(int)((thr + TB - 1) / TB), TB, 0, stream>>>(
        src, dst, dinv, T, h2, E);
    bias_act_kernel<false><<<(tot + TB - 1) / TB, TB, 0, stream>>>(h2, b2, N, GCN_OUT_DIM);
  }
}